// SumAggreg_8100308320292
// MI455X (gfx1250) — compile-verified
//
#include <hip/hip_runtime.h>

typedef __attribute__((ext_vector_type(2))) float v2f;
typedef __attribute__((ext_vector_type(8))) float v8f;

#define FEAT 128
#define GROUPS_PER_WAVE 8
#define THREADS 256

__global__ void zero_f32_kernel(float* __restrict__ p, int n) {
  int i = blockIdx.x * blockDim.x + threadIdx.x;
  if (i < n) p[i] = 0.0f;
}

// Segment sum via WMMA: per 16-row group, C[16 slots x 16 cols] += S @ X
// where S is a one-hot selection matrix from sorted segment ids.
__global__ __launch_bounds__(THREADS) void segsum_wmma_kernel(
    const float* __restrict__ mem,   // [nelems, 128] f32
    const int* __restrict__ seg2,    // int64 seg ids viewed as int pairs (low dword)
    float* __restrict__ out,         // [nseg, 128] f32 (pre-zeroed)
    int nelems) {
  const int lane = threadIdx.x & 31;
  const int m = lane & 15;       // M (slot) index for A, N (col) index for B/C
  const int half = lane >> 4;    // lane half selects K pair / M block

  const long long wave = (long long)((blockIdx.x * THREADS + threadIdx.x) >> 5);
  const long long base_row = wave * (long long)(GROUPS_PER_WAVE * 16);
  if (base_row >= nelems) return;

  // 16x16 f32 accumulator tiles, one per 16-col slab of FEAT=128
  v8f acc[8];
#pragma unroll
  for (int t = 0; t < 8; ++t)
#pragma unroll
    for (int q = 0; q < 8; ++q) acc[t][q] = 0.0f;

  int win = 0;   // segment id of slot 0
  int hi = -1;   // highest slot used since last flush (-1 = clean)

  auto wredmax = [&](int v) -> int {
#pragma unroll
    for (int d = 1; d < 16; d <<= 1) {
      int o = __shfl_xor(v, d, 32);
      v = (o > v) ? o : v;
    }
    return v;
  };
  auto wredmin = [&](int v) -> int {
#pragma unroll
    for (int d = 1; d < 16; d <<= 1) {
      int o = __shfl_xor(v, d, 32);
      v = (o < v) ? o : v;
    }
    return v;
  };

  auto flushAcc = [&]() {
    if (hi < 0) return;
#pragma unroll
    for (int j = 0; j < 8; ++j) {
      int s = j + half * 8;  // C layout: VGPR j, lane half -> M = j + 8*half
      if (s <= hi) {
        float* op = out + (long long)(win + s) * FEAT + m;
#pragma unroll
        for (int t = 0; t < 8; ++t) {
          unsafeAtomicAdd(op + t * 16, acc[t][j]);
        }
      }
    }
#pragma unroll
    for (int t = 0; t < 8; ++t)
#pragma unroll
      for (int q = 0; q < 8; ++q) acc[t][q] = 0.0f;
    hi = -1;
  };

  for (int g = 0; g < GROUPS_PER_WAVE; ++g) {
    const long long row0 = base_row + (long long)g * 16;
    if (row0 >= nelems) break;

    // lane (and its mirror in the upper half) holds seg id of row0+m
    const int segv = seg2[2 * (row0 + m)];
    const int smin = wredmin(segv);
    const int smax = wredmax(segv);

    if (g == 0) win = smin;
    if (smin - win >= 16) {  // whole group beyond current window: jump
      flushAcc();
      win = smin;
    }

    const float* rowp = mem + row0 * FEAT;

    for (;;) {
      // Build A (selection matrix) for the four K-steps under current window.
      // A 16x4 layout: lane half -> K in {0,1} vs {2,3}; .x/.y -> K even/odd.
      v2f A[4];
#pragma unroll
      for (int kb = 0; kb < 4; ++kb) {
        int sx = __shfl(segv, kb * 4 + half * 2 + 0, 32);
        int sy = __shfl(segv, kb * 4 + half * 2 + 1, 32);
        v2f a;
        a.x = (sx == win + m) ? 1.0f : 0.0f;
        a.y = (sy == win + m) ? 1.0f : 0.0f;
        A[kb] = a;
      }
      // Track highest slot actually used this pass
      int inslot = (segv >= win && segv - win < 16) ? (segv - win) : -1;
      int hp = wredmax(inslot);
      if (hp > hi) hi = hp;

      // B 4x16 layout mirrors A's K striping: lane = col, VGPR+half = row
#pragma unroll
      for (int t = 0; t < 8; ++t) {
#pragma unroll
        for (int kb = 0; kb < 4; ++kb) {
          v2f B;
          B.x = rowp[(long long)(kb * 4 + half * 2 + 0) * FEAT + t * 16 + m];
          B.y = rowp[(long long)(kb * 4 + half * 2 + 1) * FEAT + t * 16 + m];
          acc[t] = __builtin_amdgcn_wmma_f32_16x16x4_f32(
              false, A[kb], false, B, (short)0, acc[t], false, false);
        }
      }

      if (smax - win < 16) break;  // group fully covered
      // Rare: group spans beyond window. Flush and rebase to next segment run.
      flushAcc();
      int cand = (segv >= win + 16) ? segv : 0x7fffffff;
      win = wredmin(cand);
    }
  }
  flushAcc();
}

extern "C" void kernel_launch(void* const* d_in, const int* in_sizes, int n_in,
                              void* d_out, int out_size, void* d_ws, size_t ws_size,
                              hipStream_t stream) {
  const float* mem = (const float*)d_in[0];
  const int* seg2 = (const int*)d_in[1];  // int64 ids, read low dwords
  float* out = (float*)d_out;
  const int nelems = in_sizes[1];

  // 1) zero the output (harness poisons it; atomics accumulate on top)
  int zb = (out_size + 255) / 256;
  zero_f32_kernel<<<zb, 256, 0, stream>>>(out, out_size);

  // 2) windowed WMMA segment sum
  int groups = nelems / 16;  // NELEMS divisible by 16
  int waves = (groups + GROUPS_PER_WAVE - 1) / GROUPS_PER_WAVE;
  int blocks = (waves + (THREADS / 32) - 1) / (THREADS / 32);
  segsum_wmma_kernel<<<blocks, THREADS, 0, stream>>>(mem, seg2, out, nelems);
}